// HyperbolicKinematicAttention_82411832475873
// MI455X (gfx1250) — compile-verified
//
#include <hip/hip_runtime.h>
#include <hip/hip_bf16.h>

// ---------------------------------------------------------------------------
// HyperbolicKinematicAttention forward, fp32 end-to-end, CDNA5 (gfx1250).
// All GEMMs use V_WMMA_F32_16X16X4_F32 (wave32, 16x16 C tile per wave).
// Fused scores+softmax+attn@v kernel: attn tensor is written exactly once.
// ---------------------------------------------------------------------------

#define BB 32
#define JJ 512
#define EE 512
#define HH 8
#define HD 64
#define E1 513          // E + 1 (ambient dim)
#define QKVLD 1536      // 3*E
#define EPSF 1e-7f

typedef __attribute__((ext_vector_type(2))) float v2f;
typedef __attribute__((ext_vector_type(8))) float v8f;

__device__ __forceinline__ v8f wmma4(v2f a, v2f b, v8f c) {
  // D = A(16x4) * B(4x16) + C(16x16), fp32
  return __builtin_amdgcn_wmma_f32_16x16x4_f32(
      /*neg_a=*/false, a, /*neg_b=*/false, b,
      /*c_mod=*/(short)0, c, /*reuse_a=*/false, /*reuse_b=*/false);
}

// ---------------------------------------------------------------------------
// Kernel 1: x_tan = log_map0_spatial(x).  One block per (b,j) row.
// ---------------------------------------------------------------------------
__global__ void k_logmap(const float* __restrict__ x, float* __restrict__ xtan) {
  const int row = blockIdx.x;                 // 0 .. B*J-1
  const int t = threadIdx.x;                  // 0 .. 255
  const float* xr = x + (size_t)row * E1;
  __shared__ float red[256];
  float v0 = xr[1 + t];
  float v1 = xr[1 + 256 + t];
  red[t] = v0 * v0 + v1 * v1;
  __syncthreads();
  for (int off = 128; off > 0; off >>= 1) {
    if (t < off) red[t] += red[t + off];
    __syncthreads();
  }
  float nrm = sqrtf(red[0]);
  float x0 = fmaxf(xr[0], 1.0f + EPSF);
  float sc = acoshf(x0) / fmaxf(nrm, EPSF);
  float* o = xtan + (size_t)row * EE;
  o[t] = v0 * sc;
  o[256 + t] = v1 * sc;
}

// ---------------------------------------------------------------------------
// Kernel 2: qkv = x_tan @ W_qkv^T ; K columns additionally += v_tan @ W_vproj^T
// C = (16384 x 1536).  One 16x16 tile per wave, K-loop in steps of 4 (WMMA).
// ---------------------------------------------------------------------------
__global__ void k_gemm_qkv(const float* __restrict__ xtan,
                           const float* __restrict__ xvel,
                           const float* __restrict__ Wqkv,
                           const float* __restrict__ Wvp,
                           float* __restrict__ qkv) {
  const int wave = threadIdx.x >> 5;
  const int lane = threadIdx.x & 31;
  const int r = lane & 15, hf = lane >> 4;
  const int n0 = blockIdx.x * 16;                  // output column tile
  const int m0 = (blockIdx.y * 8 + wave) * 16;     // output row tile
  v8f c = {};

  const float* arow = xtan + (size_t)(m0 + r) * EE;
  const float* brow = Wqkv + (size_t)(n0 + r) * EE;   // B[k][n] = W[n][k]
  for (int k0 = 0; k0 < EE; k0 += 4) {
    v2f a, b;
    a.x = arow[k0 + 2 * hf];
    a.y = arow[k0 + 2 * hf + 1];
    b.x = brow[k0 + 2 * hf];
    b.y = brow[k0 + 2 * hf + 1];
    c = wmma4(a, b, c);
  }
  if (n0 >= EE && n0 < 2 * EE) {                   // K block: add k_vel
    const float* a2 = xvel + (size_t)(m0 + r) * E1 + 1;  // v_tan, strided rows
    const float* b2 = Wvp + (size_t)(n0 - EE + r) * EE;
    for (int k0 = 0; k0 < EE; k0 += 4) {
      v2f a, b;
      a.x = a2[k0 + 2 * hf];
      a.y = a2[k0 + 2 * hf + 1];
      b.x = b2[k0 + 2 * hf];
      b.y = b2[k0 + 2 * hf + 1];
      c = wmma4(a, b, c);
    }
  }
  float* orow = qkv + (size_t)(m0 + 8 * hf) * QKVLD + n0 + r;
#pragma unroll
  for (int i = 0; i < 8; ++i) orow[(size_t)i * QKVLD] = c[i];
}

// ---------------------------------------------------------------------------
// Kernel 3 (fused): scores (WMMA) -> +bias/scale -> row softmax -> write attn
// once -> stage P in LDS -> z_tan = P @ v (WMMA, A from LDS) -> z_manifold
// slots.  One block per (b, h, 16-query tile); 8 waves; each wave owns a
// 16x64 column slab of the 16x512 score block.
// ---------------------------------------------------------------------------
__global__ void k_attn(const float* __restrict__ qkv,
                       const float* __restrict__ bias,
                       float* __restrict__ attn,
                       float* __restrict__ zman) {
  const int wave = threadIdx.x >> 5;
  const int lane = threadIdx.x & 31;
  const int r = lane & 15, hf = lane >> 4;
  const int q0 = blockIdx.x * 16;                 // query tile
  const int z = blockIdx.y;                       // b*H + h
  const int b = z >> 3, h = z & 7;
  const int t = threadIdx.x;

  __shared__ float smem[16 * JJ];                 // 32 KB: reductions then P

  // ---- Phase 1: scores, 4 key tiles per wave (cols wave*64 .. +63) ----
  const float* qb = qkv + (size_t)(b * JJ + q0 + r) * QKVLD + h * HD;
  v2f af[16];                                     // hoisted q fragments
#pragma unroll
  for (int s = 0; s < 16; ++s) {
    af[s].x = qb[4 * s + 2 * hf];
    af[s].y = qb[4 * s + 2 * hf + 1];
  }
  v8f sc[4];
#pragma unroll
  for (int j = 0; j < 4; ++j) {
    const int n0 = wave * 64 + j * 16;
    const float* kb = qkv + (size_t)(b * JJ + n0 + r) * QKVLD + EE + h * HD;
    v8f c = {};
#pragma unroll
    for (int s = 0; s < 16; ++s) {
      v2f bb;
      bb.x = kb[4 * s + 2 * hf];
      bb.y = kb[4 * s + 2 * hf + 1];
      c = wmma4(af[s], bb, c);
    }
    sc[j] = c;
  }
  // scale + bias (C layout: VGPR i -> row i+8*hf, col n0+r)
  const float scale = 0.125f;
#pragma unroll
  for (int j = 0; j < 4; ++j) {
    const int n0 = wave * 64 + j * 16;
    const float* brow = bias + (size_t)(q0 + 8 * hf) * JJ + n0 + r;
#pragma unroll
    for (int i = 0; i < 8; ++i)
      sc[j][i] = sc[j][i] * scale + brow[(size_t)i * JJ];
  }

  // ---- Phase 2: row max over 512 cols (16 rows x 128 lane-slots) ----
#pragma unroll
  for (int i = 0; i < 8; ++i) {
    float pm = fmaxf(fmaxf(sc[0][i], sc[1][i]), fmaxf(sc[2][i], sc[3][i]));
    smem[(i + 8 * hf) * 128 + wave * 16 + r] = pm;
  }
  __syncthreads();
  for (int off = 64; off > 0; off >>= 1) {
    for (int idx = t; idx < 16 * off; idx += 256) {
      const int m = idx / off, s = idx % off;
      smem[m * 128 + s] = fmaxf(smem[m * 128 + s], smem[m * 128 + s + off]);
    }
    __syncthreads();
  }
  float rmax[8];
#pragma unroll
  for (int i = 0; i < 8; ++i) rmax[i] = smem[(i + 8 * hf) * 128];
  __syncthreads();

  // ---- Phase 3: exp, row sum, normalize, write attn once, stage P ----
#pragma unroll
  for (int j = 0; j < 4; ++j)
#pragma unroll
    for (int i = 0; i < 8; ++i) sc[j][i] = expf(sc[j][i] - rmax[i]);
#pragma unroll
  for (int i = 0; i < 8; ++i) {
    float ps = (sc[0][i] + sc[1][i]) + (sc[2][i] + sc[3][i]);
    smem[(i + 8 * hf) * 128 + wave * 16 + r] = ps;
  }
  __syncthreads();
  for (int off = 64; off > 0; off >>= 1) {
    for (int idx = t; idx < 16 * off; idx += 256) {
      const int m = idx / off, s = idx % off;
      smem[m * 128 + s] += smem[m * 128 + s + off];
    }
    __syncthreads();
  }
  float rinv[8];
#pragma unroll
  for (int i = 0; i < 8; ++i) rinv[i] = 1.0f / smem[(i + 8 * hf) * 128];
  __syncthreads();                                 // reuse smem for P

  float* arow = attn + ((size_t)z * JJ + q0 + 8 * hf) * JJ + wave * 64 + r;
#pragma unroll
  for (int j = 0; j < 4; ++j) {
#pragma unroll
    for (int i = 0; i < 8; ++i) {
      const float p = sc[j][i] * rinv[i];
      arow[(size_t)i * JJ + j * 16] = p;                       // attn output
      smem[(i + 8 * hf) * JJ + wave * 64 + j * 16 + r] = p;    // LDS stage
    }
  }
  __syncthreads();

  // ---- Phase 4: z = P(16x512) @ V(512x64); waves 0..3 own 16-col tiles ----
  if (wave < 4) {
    const float* bbase =
        qkv + (size_t)b * JJ * QKVLD + 2 * EE + h * HD + wave * 16 + r;
    v8f c = {};
    for (int k0 = 0; k0 < JJ; k0 += 4) {
      v2f a, bb;
      a.x = smem[r * JJ + k0 + 2 * hf];            // A from LDS (ds_load)
      a.y = smem[r * JJ + k0 + 2 * hf + 1];
      bb.x = bbase[(size_t)(k0 + 2 * hf) * QKVLD];
      bb.y = bbase[(size_t)(k0 + 2 * hf + 1) * QKVLD];
      c = wmma4(a, bb, c);
    }
    float* orow = zman + (size_t)(b * JJ + q0 + 8 * hf) * E1 +
                  1 + h * HD + wave * 16 + r;
#pragma unroll
    for (int i = 0; i < 8; ++i) orow[(size_t)i * E1] = c[i];
  }
}

// ---------------------------------------------------------------------------
// Kernel 4: proj = z_tan @ W_proj^T + b_proj   (16384 x 512, K=512)
//           (z_tan read from z_manifold spatial slots, ld = 513)
// ---------------------------------------------------------------------------
__global__ void k_proj(const float* __restrict__ zman,
                       const float* __restrict__ Wp,
                       const float* __restrict__ bp,
                       float* __restrict__ proj) {
  const int wave = threadIdx.x >> 5;
  const int lane = threadIdx.x & 31;
  const int r = lane & 15, hf = lane >> 4;
  const int n0 = blockIdx.x * 16;
  const int m0 = (blockIdx.y * 8 + wave) * 16;

  const float* arow = zman + (size_t)(m0 + r) * E1 + 1;
  const float* brow = Wp + (size_t)(n0 + r) * EE;
  v8f c = {};
  for (int k0 = 0; k0 < EE; k0 += 4) {
    v2f a, b;
    a.x = arow[k0 + 2 * hf];
    a.y = arow[k0 + 2 * hf + 1];
    b.x = brow[k0 + 2 * hf];
    b.y = brow[k0 + 2 * hf + 1];
    c = wmma4(a, b, c);
  }
  float biasv = bp[n0 + r];
  float* orow = proj + (size_t)(m0 + 8 * hf) * EE + n0 + r;
#pragma unroll
  for (int i = 0; i < 8; ++i) orow[(size_t)i * EE] = c[i] + biasv;
}

// ---------------------------------------------------------------------------
// Kernel 5: z_manifold = exp_map0(proj).  One block per (b,j) row.
// ---------------------------------------------------------------------------
__global__ void k_expmap(const float* __restrict__ proj, float* __restrict__ zman) {
  const int row = blockIdx.x;
  const int t = threadIdx.x;
  const float* p = proj + (size_t)row * EE;
  __shared__ float red[256];
  float v0 = p[t], v1 = p[t + 256];
  red[t] = v0 * v0 + v1 * v1;
  __syncthreads();
  for (int off = 128; off > 0; off >>= 1) {
    if (t < off) red[t] += red[t + off];
    __syncthreads();
  }
  float nrm = sqrtf(red[0]);
  float sc = sinhf(nrm) / fmaxf(nrm, EPSF);
  float* o = zman + (size_t)row * E1;
  if (t == 0) o[0] = coshf(nrm);
  o[1 + t] = v0 * sc;
  o[1 + 256 + t] = v1 * sc;
}

// ---------------------------------------------------------------------------
// Launch
// ---------------------------------------------------------------------------
extern "C" void kernel_launch(void* const* d_in, const int* in_sizes, int n_in,
                              void* d_out, int out_size, void* d_ws, size_t ws_size,
                              hipStream_t stream) {
  (void)in_sizes; (void)n_in; (void)out_size; (void)ws_size;
  const float* x    = (const float*)d_in[0];   // (B,J,E+1)
  const float* xvel = (const float*)d_in[1];   // (B,J,E+1)
  const float* bias = (const float*)d_in[2];   // (J,J)
  const float* Wqkv = (const float*)d_in[3];   // (3E,E)
  const float* Wvp  = (const float*)d_in[4];   // (E,E)
  const float* Wp   = (const float*)d_in[5];   // (E,E)
  const float* bp   = (const float*)d_in[6];   // (E,)

  float* out  = (float*)d_out;
  float* zman = out;                                         // B*J*(E+1)
  float* attn = out + (size_t)BB * JJ * E1;                  // B*H*J*J
  float* xtan = attn + (size_t)BB * HH * JJ * JJ;            // B*J*E

  float* qkv  = (float*)d_ws;   // 16384*1536 floats (100.7 MB)
  float* proj = (float*)d_ws;   // reused after attention (qkv dead)

  // 1) tangent-space projection
  k_logmap<<<dim3(BB * JJ), dim3(256), 0, stream>>>(x, xtan);
  // 2) fused qkv + velocity-key GEMM  (M=16384, N=1536, K=512 [+512 for K cols])
  k_gemm_qkv<<<dim3(QKVLD / 16, (BB * JJ) / 128), dim3(256), 0, stream>>>(
      xtan, xvel, Wqkv, Wvp, qkv);
  // 3) fused scores + softmax + attn@v (attn written exactly once)
  k_attn<<<dim3(JJ / 16, BB * HH), dim3(256), 0, stream>>>(
      qkv, bias, attn, zman);
  // 4) output projection + bias (scratch reuses qkv region)
  k_proj<<<dim3(EE / 16, (BB * JJ) / 128), dim3(256), 0, stream>>>(
      zman, Wp, bp, proj);
  // 5) exponential map back to the manifold
  k_expmap<<<dim3(BB * JJ), dim3(256), 0, stream>>>(proj, zman);
}